// smResBlock_32839319945334
// MI455X (gfx1250) — compile-verified
//
#include <hip/hip_runtime.h>
#include <hip/hip_bf16.h>
#include <math.h>
#include <stdint.h>

// Problem constants (match reference)
#define Bsz   128
#define Esz   100000
#define Fsz   5000
#define Hsz   40000
#define DIN   10
#define DOUT  10
#define EPSV  1e-5f

typedef __attribute__((ext_vector_type(2))) float v2f;
typedef __attribute__((ext_vector_type(8))) float v8f;
typedef __attribute__((ext_vector_type(4))) unsigned int u32x4;
typedef __attribute__((ext_vector_type(4))) int i32x4;
typedef __attribute__((ext_vector_type(8))) int i32x8;

#if __has_builtin(__builtin_amdgcn_tensor_load_to_lds)
#define HAVE_TDM 1
#else
#define HAVE_TDM 0
#endif

__device__ __forceinline__ float elu1(float u) {
  return u > 0.0f ? u : (__expf(u) - 1.0f);
}

// ---------------------------------------------------------------------------
// Kernel 1: xT[e,b] = x[b,e];  outT[e,b] = x[b,e] + b3[e]   (residual folded)
// ---------------------------------------------------------------------------
__global__ __launch_bounds__(256) void k_tr_in(const float* __restrict__ x,
                                               const float* __restrict__ b3,
                                               float* __restrict__ xT,
                                               float* __restrict__ outT) {
  __shared__ float tile[32][33];
  const int e0 = blockIdx.x * 32;
  const int b0 = blockIdx.y * 32;
  const int tx = threadIdx.x, ty = threadIdx.y;   // 32 x 8
#pragma unroll
  for (int yy = 0; yy < 32; yy += 8)
    tile[ty + yy][tx] = x[(long)(b0 + ty + yy) * Esz + e0 + tx];
  __syncthreads();
#pragma unroll
  for (int yy = 0; yy < 32; yy += 8) {
    const int ee = e0 + ty + yy;
    const float v = tile[tx][ty + yy];            // = x[b0+tx, ee]
    xT[(long)ee * Bsz + b0 + tx]   = v;
    outT[(long)ee * Bsz + b0 + tx] = v + b3[ee];
  }
}

// ---------------------------------------------------------------------------
// Kernel 3: out[b,e] = outT[e,b]
// ---------------------------------------------------------------------------
__global__ __launch_bounds__(256) void k_tr_out(const float* __restrict__ outT,
                                                float* __restrict__ out) {
  __shared__ float tile[32][33];
  const int e0 = blockIdx.x * 32;
  const int b0 = blockIdx.y * 32;
  const int tx = threadIdx.x, ty = threadIdx.y;
#pragma unroll
  for (int yy = 0; yy < 32; yy += 8)
    tile[ty + yy][tx] = outT[(long)(e0 + ty + yy) * Bsz + b0 + tx];
  __syncthreads();
#pragma unroll
  for (int yy = 0; yy < 32; yy += 8)
    out[(long)(b0 + ty + yy) * Esz + e0 + tx] = tile[tx][ty + yy];
}

// ---------------------------------------------------------------------------
// Kernel 2: fused  gather(spmm1)+b1 -> GLN1 -> elu(s*h)
//                  -> block-diag 8x8 mix via V_WMMA_F32_16X16X4_F32 (+b2)
//                  -> GLN2 -> elu(h*s) -> scatter(spmm3) atomics into outT
// One block (256 thr = 8 waves) = one node PAIR (16 hidden channels), all
// 128 batch rows. Wave w handles batch rows [16w, 16w+16) for WMMA; the
// epilogue re-maps threads one-per-(node, batch) so GLN2 + scatter are
// lane-local (no cross-lane reductions).
// s-gate tile is DMA'd into LDS by the Tensor Data Mover (2D strided tile,
// HW pad_interval reproduces the [128][17] anti-bank-conflict layout).
// ---------------------------------------------------------------------------
__global__ __launch_bounds__(256) void k_fused(
    const float* __restrict__ xT,   const float* __restrict__ sg,
    const float* __restrict__ w1,   const float* __restrict__ b1,
    const float* __restrict__ w2,   const float* __restrict__ b2,
    const float* __restrict__ w3,
    const float* __restrict__ g1,   const float* __restrict__ bt1,
    const float* __restrict__ g2,   const float* __restrict__ bt2,
    const int*   __restrict__ src1, const int* __restrict__ dst3,
    float* __restrict__ outT) {
  __shared__ float sL[Bsz][17];         // s[b, 16p+k], padded (17 coprime 64)
  __shared__ float zsh[2 * Bsz * 9];    // D-matrix transpose buf, stride-9 pad
  __shared__ float w1L[160];            // w1 for both nodes [node][i][c]
  __shared__ float w2L[128];            // w2 for both nodes [node][ci][co]
  __shared__ float w3L[160];            // w3 for both nodes [node][j][c]
  __shared__ float prm[6][16];          // b1,g1,bt1,b2,g2,bt2 for 16 channels
  __shared__ int   inE[20], outE[20];   // edge ids [node][i]

  const int p  = blockIdx.x;            // node pair
  const int f0 = 2 * p;
  const int t  = threadIdx.x;

  // ---- async TDM: s[0:128, 16p:16p+16] -> sL with HW row padding ----
#if HAVE_TDM
  if (t < 32) {                         // one DMA per block (wave 0 issues)
    const unsigned lds_off = (unsigned)(uintptr_t)(&sL[0][0]);
    const unsigned long long ga = (unsigned long long)(uintptr_t)(sg + 16 * p);
    u32x4 g0;
    g0[0] = 1u;                                           // count=1 (user D#)
    g0[1] = lds_off;                                      // lds_addr
    g0[2] = (unsigned)ga;                                 // global_addr[31:0]
    g0[3] = (unsigned)((ga >> 32) & 0x01FFFFFFu) | 0x80000000u; // ga[56:32]|type=2
    i32x8 g1v;
    g1v[0] = (2 << 16) | (1 << 20) | (3 << 22);           // 4B elem, pad_en,
                                                          // every 16 dw, +1 dw
    g1v[1] = (int)(16u << 16);                            // tensor_dim0 = 16
    g1v[2] = (int)(128u << 16);                           // tensor_dim1 = 128
    g1v[3] = (int)(16u << 16);                            // tile_dim0 = 16
    g1v[4] = 128;                                         // tile_dim1 = 128
    g1v[5] = Hsz;                                         // dim0_stride = 40000
    g1v[6] = 0; g1v[7] = 0;
    i32x4 gz = {0, 0, 0, 0};                              // 2D: groups 2/3 = 0
#if defined(__clang_major__) && (__clang_major__ >= 23)
    i32x8 gz8 = {0, 0, 0, 0, 0, 0, 0, 0};
    __builtin_amdgcn_tensor_load_to_lds(g0, g1v, gz, gz, gz8, 0);
#else
    __builtin_amdgcn_tensor_load_to_lds(g0, g1v, gz, gz, 0);
#endif
    __builtin_amdgcn_s_wait_tensorcnt(0);
  }
#endif

  // ---- cooperative staging into LDS (overlaps with TDM) ----
  if (t < 20)  inE[t]  = src1[((f0 + t / DIN) * DIN + (t % DIN)) * 8];
  if (t >= 32 && t < 52) {
    int q = t - 32;
    outE[q] = dst3[((f0 + q / DOUT) * DOUT + (q % DOUT)) * 8];
  }
  if (t < 160) { w1L[t] = w1[f0 * 80 + t]; w3L[t] = w3[f0 * 80 + t]; }
  if (t < 128) w2L[t] = w2[f0 * 64 + t];
  if (t >= 128 && t < 224) {
    int j = (t - 128) >> 4, k = t & 15;
    const float* srcs[6] = {b1, g1, bt1, b2, g2, bt2};
    prm[j][k] = srcs[j][16 * p + k];
  }
#if !HAVE_TDM
  for (int idx = t; idx < Bsz * 16; idx += 256)
    sL[idx >> 4][idx & 15] = sg[(long)(idx >> 4) * Hsz + 16 * p + (idx & 15)];
#endif
  __syncthreads();

  const int lane = t & 31;
  const int wave = t >> 5;              // 0..7 (batch tile)
  const int bl   = lane & 15;           // batch-local / N column
  const int half = lane >> 4;           // 0/1: node select (stage 1)
  const int b    = wave * 16 + bl;      // global batch row (stage 1)

  // ---- stage 1: gather + GLN1 + elu(s*h); lane owns (b, node f0+half) ----
  float xv[DIN];
#pragma unroll
  for (int i = 0; i < DIN; ++i)
    xv[i] = xT[(long)inE[half * DIN + i] * Bsz + b];   // coalesced 64B/half

  float a[8];
#pragma unroll
  for (int c = 0; c < 8; ++c) {
    float acc = prm[0][half * 8 + c];
#pragma unroll
    for (int i = 0; i < DIN; ++i)
      acc = fmaf(xv[i], w1L[half * 80 + i * 8 + c], acc);
    a[c] = acc;
  }
  { // lane-local group norm over the 8 channels
    float sm = 0.f, sq = 0.f;
#pragma unroll
    for (int c = 0; c < 8; ++c) { sm += a[c]; sq += a[c] * a[c]; }
    const float mu  = sm * 0.125f;
    const float var = fmaf(-mu, mu, sq * 0.125f);
    const float inv = rsqrtf(var + EPSV);
#pragma unroll
    for (int c = 0; c < 8; ++c) {
      float y = fmaf(prm[1][half * 8 + c], (a[c] - mu) * inv, prm[2][half * 8 + c]);
      a[c] = elu1(y * sL[b][half * 8 + c]);
    }
  }

  // ---- stage 2: 16x16 block-diag GEMM via 4x V_WMMA_F32_16X16X4_F32 ----
  // B (4x16) slice j: lanes<16 hold rows {4j,4j+1}, lanes>=16 rows {4j+2,4j+3}
  v2f Bm[4];
#pragma unroll
  for (int j = 0; j < 4; ++j) {
    const int k0 = 4 * j + half * 2;
    const int k1 = k0 + 1;
    Bm[j].x = ((k0 >> 3) == (bl >> 3)) ? w2L[(k0 >> 3) * 64 + (k0 & 7) * 8 + (bl & 7)] : 0.f;
    Bm[j].y = ((k1 >> 3) == (bl >> 3)) ? w2L[(k1 >> 3) * 64 + (k1 & 7) * 8 + (bl & 7)] : 0.f;
  }

  v8f acc = {};
#pragma unroll
  for (int j = 0; j < 4; ++j) {
    // A (16x4) slice j: source (b=bl, k=4j..4j+3); k<8 lives in lanes 0-15,
    // k>=8 in lanes 16-31, channel = k&7 -> register a[(j&1)*4 + ..]
    const int srcLane = bl + ((j >> 1) << 4);
    const int cc = (j & 1) * 4;
    const float s0 = __shfl(a[cc + 0], srcLane, 32);
    const float s1 = __shfl(a[cc + 1], srcLane, 32);
    const float s2 = __shfl(a[cc + 2], srcLane, 32);
    const float s3 = __shfl(a[cc + 3], srcLane, 32);
    v2f Am;
    Am.x = half ? s2 : s0;   // lanes 0-15: K=4j ; lanes 16-31: K=4j+2
    Am.y = half ? s3 : s1;   // lanes 0-15: K=4j+1; lanes 16-31: K=4j+3
    acc = __builtin_amdgcn_wmma_f32_16x16x4_f32(
        false, Am, false, Bm[j], (short)0, acc, false, false);
  }

  // ---- transpose D through LDS: lane (vgpr r) -> element (b, node, c) ----
  // D layout: vgpr r, lane: n=lane&15, batch = wave*16 + r + 8*(lane>>4).
#pragma unroll
  for (int r = 0; r < 8; ++r) {
    const int bb = wave * 16 + r + 8 * half;
    zsh[((bl >> 3) * Bsz + bb) * 9 + (bl & 7)] = acc[r] + prm[3][bl];
  }
  __syncthreads();

  // ---- epilogue: one thread per (node, batch): lane-local GLN2 + elu,
  //      then the 10 output-edge dot products + coalesced atomics ----
  {
    const int node = t >> 7;            // 0/1
    const int bq   = t & 127;           // batch row
    float z[8];
    float sm = 0.f, sq = 0.f;
#pragma unroll
    for (int c = 0; c < 8; ++c) {
      z[c] = zsh[(node * Bsz + bq) * 9 + c];
      sm += z[c]; sq += z[c] * z[c];
    }
    const float mu  = sm * 0.125f;
    const float var = fmaf(-mu, mu, sq * 0.125f);
    const float inv = rsqrtf(var + EPSV);
#pragma unroll
    for (int c = 0; c < 8; ++c) {
      float y = fmaf(prm[4][node * 8 + c], (z[c] - mu) * inv, prm[5][node * 8 + c]);
      z[c] = elu1(y * sL[bq][node * 8 + c]);
    }
#pragma unroll
    for (int j = 0; j < DOUT; ++j) {
      float acc3 = 0.f;
#pragma unroll
      for (int c = 0; c < 8; ++c)
        acc3 = fmaf(z[c], w3L[node * 80 + j * 8 + c], acc3);
      atomicAdd(outT + (long)outE[node * DOUT + j] * Bsz + bq, acc3);
    }
  }
}

// ---------------------------------------------------------------------------
extern "C" void kernel_launch(void* const* d_in, const int* in_sizes, int n_in,
                              void* d_out, int out_size, void* d_ws, size_t ws_size,
                              hipStream_t stream) {
  (void)in_sizes; (void)n_in; (void)out_size; (void)ws_size;
  const float* x    = (const float*)d_in[0];
  const float* s    = (const float*)d_in[1];
  const float* w1   = (const float*)d_in[2];
  const float* b1   = (const float*)d_in[3];
  const float* w2   = (const float*)d_in[4];
  const float* b2   = (const float*)d_in[5];
  const float* w3   = (const float*)d_in[6];
  const float* b3   = (const float*)d_in[7];
  const float* g1   = (const float*)d_in[8];
  const float* bt1  = (const float*)d_in[9];
  const float* g2   = (const float*)d_in[10];
  const float* bt2  = (const float*)d_in[11];
  const int*   src1 = (const int*)d_in[12];
  const int*   dst3 = (const int*)d_in[17];
  float* out = (float*)d_out;

  // workspace: xT[E,B] then outT[E,B]  (2 * 51.2 MB; L2-resident on MI455X)
  float* xT   = (float*)d_ws;
  float* outT = xT + (size_t)Esz * Bsz;

  dim3 tb(32, 8);
  dim3 tg(Esz / 32, Bsz / 32);
  k_tr_in<<<tg, tb, 0, stream>>>(x, b3, xT, outT);
  k_fused<<<Fsz / 2, 256, 0, stream>>>(xT, s, w1, b1, w2, b2, w3,
                                       g1, bt1, g2, bt2, src1, dst3, outT);
  k_tr_out<<<tg, tb, 0, stream>>>(outT, out);
}